// neko_DTDNG_mk1_30322469109878
// MI455X (gfx1250) — compile-verified
//
#include <hip/hip_runtime.h>

typedef __attribute__((ext_vector_type(2))) float v2f;
typedef __attribute__((ext_vector_type(8))) float v8f;

#define NB 64
#define NC 768
#define NH 16
#define NW 64
#define NT 30
#define NP 4
#define AH 8
#define AW 32
#define M_REAL 120          // NT*NP
#define M_PAD  128
#define KDIM   1024         // NH*NW
#define KC     32
#define LDA    36           // KC + 4 pad; 36*4B = 144B keeps 16B alignment per row
#define NCHUNK (KDIM / KC)  // 32

// ---------------------------------------------------------------------------
// Kernel 1: bilinear resize (8x32 -> 16x64) + L1 normalization (sum + 1e-4),
// writes padded RA[b][128][1024] into workspace (rows 120..127 zero).
// One 256-thread block per (b, tp).
// ---------------------------------------------------------------------------
__global__ __launch_bounds__(256) void neko_resize_norm(
    const float* __restrict__ A, float* __restrict__ RA) {
  const int bidx = blockIdx.x;
  const int b  = bidx >> 7;
  const int tp = bidx & 127;
  const int tid = threadIdx.x;
  float* __restrict__ row_out = RA + ((size_t)b * M_PAD + tp) * KDIM;

  if (tp >= M_REAL) {   // uniform per block: pad rows -> zeros
    float4 z = make_float4(0.f, 0.f, 0.f, 0.f);
    *(float4*)(row_out + tid * 4) = z;
    return;
  }

  __shared__ float sIn[AH * AW];   // 256 == blockDim
  __shared__ float sRed[256];

  const int t = tp >> 2;
  const int p = tp & 3;
  sIn[tid] = A[(((size_t)b * NT + t) * NP + p) * (AH * AW) + tid];
  __syncthreads();

  float vals[4];
  float lsum = 0.f;
#pragma unroll
  for (int j = 0; j < 4; ++j) {
    const int hw = tid * 4 + j;
    const int oy = hw >> 6;
    const int ox = hw & 63;
    // scale = in/out = 0.5 for both axes: src = (o+0.5)*0.5 - 0.5
    float sy = fmaxf(0.5f * (float)oy - 0.25f, 0.f);
    int y0 = (int)sy; y0 = y0 > (AH - 1) ? (AH - 1) : y0;
    int y1 = (y0 + 1 > AH - 1) ? (AH - 1) : (y0 + 1);
    float wy = sy - (float)y0;

    float sx = fmaxf(0.5f * (float)ox - 0.25f, 0.f);
    int x0 = (int)sx; x0 = x0 > (AW - 1) ? (AW - 1) : x0;
    int x1 = (x0 + 1 > AW - 1) ? (AW - 1) : (x0 + 1);
    float wx = sx - (float)x0;

    const float v00 = sIn[y0 * AW + x0];
    const float v01 = sIn[y0 * AW + x1];
    const float v10 = sIn[y1 * AW + x0];
    const float v11 = sIn[y1 * AW + x1];
    const float v = (v00 * (1.f - wy) + v10 * wy) * (1.f - wx) +
                    (v01 * (1.f - wy) + v11 * wy) * wx;
    vals[j] = v;
    lsum += v;
  }

  sRed[tid] = lsum;
  __syncthreads();
#pragma unroll
  for (int s = 128; s > 0; s >>= 1) {
    if (tid < s) sRed[tid] += sRed[tid + s];
    __syncthreads();
  }
  const float inv = 1.f / (sRed[0] + 1e-4f);
  float4 o = make_float4(vals[0] * inv, vals[1] * inv, vals[2] * inv, vals[3] * inv);
  *(float4*)(row_out + tid * 4) = o;
}

// ---------------------------------------------------------------------------
// Async global -> LDS copy (CDNA5), tracked by ASYNCcnt.
// lds_off: AS(3) byte offset (low 32 bits of the generic pointer into shared).
// ---------------------------------------------------------------------------
__device__ __forceinline__ void async_copy_b128(const float* gptr, const float* lds_ptr) {
  const unsigned lds_off = (unsigned)(unsigned long long)(uintptr_t)lds_ptr;
  asm volatile("global_load_async_to_lds_b128 %0, %1, off"
               :: "v"(lds_off), "v"((unsigned long long)(uintptr_t)gptr)
               : "memory");
}

// Issue the 6 async b128 loads for one K-chunk (4 for RA rows, 2 for feature).
__device__ __forceinline__ void stage_chunk(const float* __restrict__ gA,
                                            const float* __restrict__ gB,
                                            float* sAbuf, float* sBbuf,
                                            int tid, int k0) {
#pragma unroll
  for (int i = 0; i < 4; ++i) {          // A: 128 rows x 8 float4 = 1024, 4/thread
    const int idx = tid + 256 * i;
    const int row = idx >> 3;
    const int c4  = idx & 7;
    async_copy_b128(gA + (size_t)row * KDIM + k0 + c4 * 4, &sAbuf[row * LDA + c4 * 4]);
  }
#pragma unroll
  for (int i = 0; i < 2; ++i) {          // B: 64 ch x 8 float4 = 512, 2/thread
    const int idx = tid + 256 * i;
    const int ch = idx >> 3;
    const int c4 = idx & 7;
    async_copy_b128(gB + (size_t)ch * KDIM + k0 + c4 * 4, &sBbuf[ch * LDA + c4 * 4]);
  }
}

// ---------------------------------------------------------------------------
// Kernel 2: per-batch GEMM  C[b](120x768) = RA[b](128x1024) x F[b]^T(1024x768)
// V_WMMA_F32_16X16X4_F32, double-buffered LDS fed by async-to-LDS copies.
// Block = 256 threads = 8 waves; wave w owns M-tile w x four 16x16 N-tiles.
// ---------------------------------------------------------------------------
__global__ __launch_bounds__(256) void neko_wmma_gemm(
    const float* __restrict__ F, const float* __restrict__ RA,
    float* __restrict__ C) {
  const int b   = blockIdx.x;
  const int n0  = blockIdx.y * 64;
  const int tid = threadIdx.x;
  const int wave = tid >> 5;
  const int lane = tid & 31;
  const int lrow  = lane & 15;
  const int lhalf = (lane >> 4) << 1;   // 0 for lanes 0-15, 2 for lanes 16-31

  __shared__ float sA[2][M_PAD * LDA];  // 2 x 18 KB
  __shared__ float sB[2][64 * LDA];     // 2 x  9 KB

  const float* __restrict__ gA = RA + (size_t)b * M_PAD * KDIM;
  const float* __restrict__ gB = F + ((size_t)b * NC + n0) * KDIM;

  v8f acc[4];
#pragma unroll
  for (int i = 0; i < 4; ++i)
#pragma unroll
    for (int j = 0; j < 8; ++j) acc[i][j] = 0.f;

  // Prologue: fill buffer 0.
  stage_chunk(gA, gB, sA[0], sB[0], tid, 0);

  for (int it = 0; it < NCHUNK; ++it) {
    const int buf = it & 1;
    if (it + 1 < NCHUNK) {
      // Prefetch next chunk into the other buffer, then wait only for the
      // oldest 6 async loads (this chunk) -- async loads complete in order.
      stage_chunk(gA, gB, sA[buf ^ 1], sB[buf ^ 1], tid, (it + 1) * KC);
      asm volatile("s_wait_asynccnt 0x6" ::: "memory");
    } else {
      asm volatile("s_wait_asynccnt 0x0" ::: "memory");
    }
    __syncthreads();   // all waves' chunk-it data resident in LDS

    const float* __restrict__ cA = sA[buf];
    const float* __restrict__ cB = sB[buf];
#pragma unroll
    for (int kk = 0; kk < KC; kk += 4) {
      // A-matrix 16x4 fragment (lanes 0-15: K=kk,kk+1; lanes 16-31: K=kk+2,kk+3)
      v2f a = *(const v2f*)(&cA[(wave * 16 + lrow) * LDA + kk + lhalf]);
#pragma unroll
      for (int nt = 0; nt < 4; ++nt) {
        // B-matrix 4x16 fragment: B[k][n] = feature[n][k], mirrored lane layout
        v2f bb = *(const v2f*)(&cB[(nt * 16 + lrow) * LDA + kk + lhalf]);
        acc[nt] = __builtin_amdgcn_wmma_f32_16x16x4_f32(
            false, a, false, bb, (short)0, acc[nt], false, false);
      }
    }
    __syncthreads();   // all waves done reading buf -> safe to refill at it+2
  }

  // D layout: VGPR r -> row r (lanes 0-15) / row 8+r (lanes 16-31), col = lane&15
#pragma unroll
  for (int nt = 0; nt < 4; ++nt) {
#pragma unroll
    for (int r = 0; r < 8; ++r) {
      const int row = wave * 16 + r + ((lane >> 4) << 3);
      if (row < M_REAL) {
        C[((size_t)b * M_REAL + row) * NC + n0 + nt * 16 + lrow] = acc[nt][r];
      }
    }
  }
}

extern "C" void kernel_launch(void* const* d_in, const int* in_sizes, int n_in,
                              void* d_out, int out_size, void* d_ws, size_t ws_size,
                              hipStream_t stream) {
  (void)in_sizes; (void)n_in; (void)out_size; (void)ws_size;
  const float* feature = (const float*)d_in[0];   // (64,768,16,64) fp32
  const float* A       = (const float*)d_in[1];   // (64,30,4,8,32) fp32
  // d_in[2] = text_length (int scalar) == NT in this fixed configuration.
  float* out = (float*)d_out;                     // (64,30,4,768) fp32
  float* RA  = (float*)d_ws;                      // padded (64,128,1024) fp32 = 33.6 MB

  neko_resize_norm<<<NB * M_PAD, 256, 0, stream>>>(A, RA);
  neko_wmma_gemm<<<dim3(NB, NC / 64), 256, 0, stream>>>(feature, RA, out);
}